// FFTLongConv_32066225832671
// MI455X (gfx1250) — compile-verified
//
#include <hip/hip_runtime.h>
#include <hip/hip_bf16.h>

// FFT long conv == odd-length (8193) circular convolution, computed exactly in
// the time domain as block-Toeplitz WMMA matmuls (f32 16x16x4), + bias + mask.
// Wave register tile: 2 batches x 2 output blocks -> 4 WMMAs per A/B fetch.

typedef __attribute__((ext_vector_type(2))) float v2f;
typedef __attribute__((ext_vector_type(4))) float v4f;
typedef __attribute__((ext_vector_type(8))) float v8f;

#define B_   4
#define H_   16
#define D_   64
#define L_   8192
#define N_   8193      // circular length (L+1)
#define CJ   512       // j-range per LDS chunk (128 WMMA steps, K=4 each)
#define NITER (CJ/4)
#define FCS  2336      // staged filter slice: spans 8 output blocks: 511+1792+16 -> 2319, padded
#define XSS  768       // staged x slice per batch: CJ + 242, padded
#define NCHUNK 17
#define JSTART (-256)  // j0 sweeps [-256, 8188] step 4 (covers all taps for n=0..15)

#define WMMA_F32(Aa, Bb, Cc) \
    __builtin_amdgcn_wmma_f32_16x16x4_f32(false, (Aa), false, (Bb), (short)0, (Cc), false, false)

__global__ __launch_bounds__(256)
void fftconv_wmma_kernel(const float* __restrict__ x,     // (B,H,D,L)
                         const float* __restrict__ f,     // (H,D,L)
                         const float* __restrict__ bias,  // (D)
                         const int*  __restrict__ pos,    // (B,L)
                         float* __restrict__ out)         // (B,H,D,L)
{
    __shared__ float fcs[FCS];        // wrapped-filter Toeplitz slice (shared by all waves)
    __shared__ float xs[B_][XSS];     // zero-padded input slices, one per batch

    const int mgrp = blockIdx.x;      // 0..3 : which 2048-output group
    const int hd   = blockIdx.y;      // 0..1023: (h*64 + d)
    const int d    = hd & (D_ - 1);
    const int mb0  = mgrp * 2048;

    const int tid  = threadIdx.x;
    const int lane = tid & 31;
    const int wave = tid >> 5;        // 8 waves
    const int bp   = wave & 1;        // batch pair: batches {2bp, 2bp+1}
    const int wq   = wave >> 1;       // block pair: blocks {2wq, 2wq+1}
    const int b0   = 2 * bp, b1 = 2 * bp + 1;

    const float* __restrict__ frow = f + (size_t)hd * L_;

    // Wave-invariant WMMA operand indices (wave32 VGPR layouts, ISA 7.12.2):
    //  A (16x4 f32): lane<16 -> {v0=fc[D+M], v1=fc[D+M-1]}; lane>=16 -> K=2,3 at M=lane-16.
    const int off   = (lane < 16) ? lane : (lane - 18);
    const int aidx0 = 512 * wq + (CJ - 1) + off;   // fcs index, K-even elem, block 2wq, t=0
    //  B (4x16 f32): lane<16 holds rows K=0,1 at N=lane; lane>=16 rows K=2,3 at N=lane-16.
    const int bidx0 = 16 * (lane & 15) + ((lane >> 4) << 1);

    v8f acc00 = {}, acc01 = {}, acc10 = {}, acc11 = {};   // [batch][block]

    for (int c = 0; c < NCHUNK; ++c) {
        const int jc = JSTART + c * CJ;
        __syncthreads();   // previous chunk's readers done before restaging

        // ---- stage wrapped filter: fcs[s] = fc(fc_lo + s), with N=8193 wrap ----
        const int fc_lo = mb0 - jc - (CJ - 1);
        for (int s = tid; s < FCS; s += 256) {
            int tt = fc_lo + s;
            float v = 0.0f;
            if (tt >= 0) { if (tt < L_) v = frow[tt]; }
            else         { int u = tt + N_; if (u >= 2 && u < L_) v = frow[u]; }
            fcs[s] = v;
        }

        // ---- stage x for all 4 batches ----
        if (jc >= 0 && jc + XSS <= L_) {
            // Interior chunk: window fully in-range -> async B128 copies to LDS
            // (GLOBAL_LOAD_ASYNC_TO_LDS_B128, GV mode, tracked by ASYNCcnt).
            #pragma unroll
            for (int q = 0; q < 3; ++q) {
                int slot = tid + q * 256;            // 768 float4 slots over 4 rows
                int bb   = slot / 192;               // 192 float4 per row
                int ss4  = (slot - bb * 192) * 4;    // float index within row
                const float* gsrc = x + (size_t)(bb * (H_ * D_) + hd) * L_ + jc + ss4;
                unsigned ldsoff = (unsigned)(uintptr_t)&xs[bb][ss4];
                unsigned long long ga = (unsigned long long)(uintptr_t)gsrc;
                asm volatile("global_load_async_to_lds_b128 %0, %1, off"
                             :: "v"(ldsoff), "v"(ga) : "memory");
            }
        } else {
            // Edge chunk: guarded loads implement the zero padding.
            for (int bb = 0; bb < B_; ++bb) {
                const float* __restrict__ xrow = x + (size_t)(bb * (H_ * D_) + hd) * L_;
                for (int ss = tid; ss < XSS; ss += 256) {
                    int j = jc + ss;
                    xs[bb][ss] = ((unsigned)j < (unsigned)L_) ? xrow[j] : 0.0f;
                }
            }
        }

        // ---- prefetch next chunk's fresh regions (global_prefetch_b8) ----
        if (c + 1 < NCHUNK) {
            const int jn = jc + CJ;
            if (tid < 128) {
                int bb = tid >> 5, q = tid & 31;
                int j  = jn + 240 + q * 16;                 // fresh x region next chunk
                if ((unsigned)j < (unsigned)L_) {
                    const float* xrow = x + (size_t)(bb * (H_ * D_) + hd) * L_;
                    __builtin_prefetch(xrow + j, 0, 1);
                }
            } else {
                int q  = tid - 128;
                int tt = fc_lo - CJ + q * 4;                // fc window slides down by CJ
                int u  = (tt >= 0) ? tt : tt + N_;
                if ((unsigned)u < (unsigned)L_) __builtin_prefetch(frow + u, 0, 1);
            }
        }

        asm volatile("s_wait_asynccnt 0x0" ::: "memory");   // async LDS writes landed
        __syncthreads();

        // ---- compute: 4 WMMAs per A/B fetch, wave-uniform bounds (EXEC all-1) ----
        int niter = (8188 - jc) >> 2;          // last valid t with j0 = jc + 4t <= 8188
        if (niter > NITER - 1) niter = NITER - 1;
        int aidx = aidx0;
        int bidx = bidx0;
        const float* __restrict__ xsb0 = xs[b0];
        const float* __restrict__ xsb1 = xs[b1];
#pragma unroll 2
        for (int t = 0; t <= niter; ++t) {
            v2f A0, A1, Bv0, Bv1;
            A0.x  = fcs[aidx];         A0.y  = fcs[aidx - 1];     // block 2wq
            A1.x  = fcs[aidx + 256];   A1.y  = fcs[aidx + 255];   // block 2wq+1
            Bv0.x = xsb0[bidx];        Bv0.y = xsb0[bidx + 1];
            Bv1.x = xsb1[bidx];        Bv1.y = xsb1[bidx + 1];
            acc00 = WMMA_F32(A0, Bv0, acc00);   // 4 independent accumulation chains
            acc01 = WMMA_F32(A1, Bv0, acc01);
            acc10 = WMMA_F32(A0, Bv1, acc10);
            acc11 = WMMA_F32(A1, Bv1, acc11);
            aidx -= 4;                 // Toeplitz: filter window slides down
            bidx += 4;                 // x window slides up
        }
    }

    // ---- epilogue: 1/N scale + x*bias + position mask; 2x float4 per lane/tile ----
    const int   col   = lane & 15;
    const int   half  = lane >> 4;
    const float bv    = bias[d];
    const float inv_n = 1.0f / 8193.0f;

    auto store_tile = [&](const v8f& a, int b, int mb) {
        const int m0 = mb + 16 * col + 8 * half;   // C layout -> 8 contiguous m per lane
        const size_t rowbase = (size_t)(b * (H_ * D_) + hd) * L_;
        const int* __restrict__ prow = pos + (size_t)b * L_;
        float res[8];
        #pragma unroll
        for (int r = 0; r < 8; ++r) {
            float xv = x[rowbase + m0 + r];
            float v  = a[r] * inv_n + xv * bv;
            res[r]   = (prow[m0 + r] != -1) ? v : 0.0f;
        }
        v4f lo = { res[0], res[1], res[2], res[3] };
        v4f hi = { res[4], res[5], res[6], res[7] };
        *(v4f*)(out + rowbase + m0)     = lo;
        *(v4f*)(out + rowbase + m0 + 4) = hi;
    };

    const int mbA = mb0 + 512 * wq;        // block 2wq
    const int mbB = mbA + 256;             // block 2wq+1
    store_tile(acc00, b0, mbA);
    store_tile(acc01, b0, mbB);
    store_tile(acc10, b1, mbA);
    store_tile(acc11, b1, mbB);
}

extern "C" void kernel_launch(void* const* d_in, const int* in_sizes, int n_in,
                              void* d_out, int out_size, void* d_ws, size_t ws_size,
                              hipStream_t stream) {
    (void)in_sizes; (void)n_in; (void)out_size; (void)d_ws; (void)ws_size;
    const float* x    = (const float*)d_in[0];   // (4,16,64,8192) f32
    const float* f    = (const float*)d_in[1];   // (16,64,8192)   f32
    const float* bias = (const float*)d_in[2];   // (1,1,64,1)     f32
    const int*   pos  = (const int*)  d_in[3];   // (4,8192)       i32
    float*       out  = (float*)d_out;

    dim3 grid(4, H_ * D_, 1);    // 4 groups of 2048 outputs x 1024 (h,d) channels
    dim3 block(256, 1, 1);       // 8 wave32: {2 batch-pairs} x {4 block-pairs}
    fftconv_wmma_kernel<<<grid, block, 0, stream>>>(x, f, bias, pos, out);
}